// Net_33397665694120
// MI455X (gfx1250) — compile-verified
//
#include <hip/hip_runtime.h>
#include <math.h>

typedef __attribute__((ext_vector_type(2))) float v2f;
typedef __attribute__((ext_vector_type(8))) float v8f;

// ---------------------------------------------------------------- atomics
__device__ __forceinline__ void atomAddF(float* p, float v) {
    unsafeAtomicAdd(p, v);   // hardware global_atomic_add_f32
}

// ---------------------------------------------------------------- init deg = 1.0 (self-loop)
__global__ void __launch_bounds__(256) k_set_one(float* __restrict__ deg, int n) {
    int i = blockIdx.x * blockDim.x + threadIdx.x;
    if (i < n) deg[i] = 1.0f;
}

// ---------------------------------------------------------------- deg accumulation over dst
__global__ void __launch_bounds__(256) k_degree(const int* __restrict__ dst,
                                                float* __restrict__ deg, int E) {
    int e = blockIdx.x * blockDim.x + threadIdx.x;
    if (e < E) atomAddF(&deg[(unsigned)dst[e]], 1.0f);
}

// ---------------------------------------------------------------- dinv = rsqrt(deg) in place
__global__ void __launch_bounds__(256) k_rsqrt(float* __restrict__ d, int n) {
    int i = blockIdx.x * blockDim.x + threadIdx.x;
    if (i < n) {
        float v = d[i];
        d[i] = (v > 0.0f) ? __frsqrt_rn(v) : 0.0f;
    }
}

// ---------------------------------------------------------------- layer1 GEMM: h1[N,16] = x[N,21] @ W1[21,16]
// one wave per 16-node tile; V_WMMA_F32_16X16X4_F32; K = 21 -> 5 full chunks + masked tail
__global__ void __launch_bounds__(128) k_gemm1(const float* __restrict__ x,
                                               const float* __restrict__ W1,
                                               float* __restrict__ h1, int nNodes) {
    int lane = threadIdx.x & 31;
    int wave = threadIdx.x >> 5;
    int tile = blockIdx.x * (blockDim.x >> 5) + wave;
    int nb = tile * 16;
    if (nb >= nNodes) return;

    int m  = lane & 15;                 // A row / B,D column
    int hi = lane >> 4;                 // half-wave K-pair select
    int node = nb + m;
    if (node >= nNodes) node = nNodes - 1;   // clamp (extra rows never stored)
    const float* xr = x + (unsigned)node * 21u;

    v8f c = {};
    #pragma unroll
    for (int kb = 0; kb < 20; kb += 4) {     // k in [0,20): all valid, branchless
        int k0 = kb + 2 * hi;
        v2f a, b;
        a.x = xr[(unsigned)k0];
        a.y = xr[(unsigned)(k0 + 1)];
        b.x = W1[(unsigned)(k0 * 16 + m)];
        b.y = W1[(unsigned)((k0 + 1) * 16 + m)];
        c = __builtin_amdgcn_wmma_f32_16x16x4_f32(false, a, false, b, (short)0, c, false, false);
    }
    {   // tail chunk kb=20: only k==20 (hi==0, first element) is real
        float msk = hi ? 0.0f : 1.0f;
        int  k0c  = hi ? 0 : 20;             // clamped safe address
        v2f a, b;
        a.x = xr[(unsigned)k0c] * msk;              a.y = 0.0f;
        b.x = W1[(unsigned)(k0c * 16 + m)] * msk;   b.y = 0.0f;
        c = __builtin_amdgcn_wmma_f32_16x16x4_f32(false, a, false, b, (short)0, c, false, false);
    }

    unsigned base = (unsigned)(nb + 8 * hi) * 16u + (unsigned)m;
    if (nb + 16 <= nNodes) {            // fast path: whole tile in range (uniform branch)
        #pragma unroll
        for (int r = 0; r < 8; ++r) h1[base + (unsigned)r * 16u] = c[r];
    } else {
        #pragma unroll
        for (int r = 0; r < 8; ++r) {
            int row = nb + r + 8 * hi;
            if (row < nNodes) h1[(unsigned)row * 16u + (unsigned)m] = c[r];
        }
    }
}

// ---------------------------------------------------------------- layer1 edge scatter: 16 lanes per edge
__global__ void __launch_bounds__(256) k_edge1(const int* __restrict__ src,
                                               const int* __restrict__ dst,
                                               const float* __restrict__ dinv,
                                               const float* __restrict__ h1,
                                               float* __restrict__ agg1, int E) {
    int t = blockIdx.x * blockDim.x + threadIdx.x;
    int e = t >> 4;
    unsigned f = (unsigned)(t & 15);
    if (e >= E) return;
    unsigned s = (unsigned)src[e], d = (unsigned)dst[e];  // broadcast within 16 lanes
    float w = dinv[s] * dinv[d];
    atomAddF(agg1 + d * 16u + f, h1[s * 16u + f] * w);    // coalesced 64B burst
}

// ---------------------------------------------------------------- layer1 finalize: self-loop + bias + relu (in place)
__global__ void __launch_bounds__(256) k_fin1(float* __restrict__ agg1,
                                              const float* __restrict__ h1,
                                              const float* __restrict__ dinv,
                                              const float* __restrict__ b1, int n) {
    int t = blockIdx.x * blockDim.x + threadIdx.x;
    if (t >= n * 16) return;
    unsigned u = (unsigned)t;
    unsigned i = u >> 4, f = u & 15u;
    float di = dinv[i];
    float v = agg1[u] + h1[u] * di * di + b1[f];
    agg1[u] = v > 0.0f ? v : 0.0f;
}

// ---------------------------------------------------------------- layer2 GEMM: h2[N,6] = h1r[N,16] @ W2[16,6]
__global__ void __launch_bounds__(128) k_gemm2(const float* __restrict__ h,
                                               const float* __restrict__ W2,
                                               float* __restrict__ h2, int nNodes) {
    int lane = threadIdx.x & 31;
    int wave = threadIdx.x >> 5;
    int tile = blockIdx.x * (blockDim.x >> 5) + wave;
    int nb = tile * 16;
    if (nb >= nNodes) return;

    int m  = lane & 15;
    int hi = lane >> 4;
    int node = nb + m;
    if (node >= nNodes) node = nNodes - 1;
    const float2* hr = (const float2*)(h + (unsigned)node * 16u);   // 8B-aligned b64 loads

    float bm = (m < 6) ? 1.0f : 0.0f;
    int   mc = (m < 6) ? m : 0;                                     // clamped, branchless

    v8f c = {};
    #pragma unroll
    for (int kb = 0; kb < 16; kb += 4) {
        int k0 = kb + 2 * hi;
        float2 av = hr[(unsigned)(k0 >> 1)];
        v2f a, b;
        a.x = av.x;
        a.y = av.y;
        b.x = W2[(unsigned)(k0 * 6 + mc)] * bm;
        b.y = W2[(unsigned)((k0 + 1) * 6 + mc)] * bm;
        c = __builtin_amdgcn_wmma_f32_16x16x4_f32(false, a, false, b, (short)0, c, false, false);
    }

    if (m < 6) {
        unsigned base = (unsigned)(nb + 8 * hi) * 6u + (unsigned)m;
        if (nb + 16 <= nNodes) {        // fast path (uniform)
            #pragma unroll
            for (int r = 0; r < 8; ++r) h2[base + (unsigned)r * 6u] = c[r];
        } else {
            #pragma unroll
            for (int r = 0; r < 8; ++r) {
                int row = nb + r + 8 * hi;
                if (row < nNodes) h2[(unsigned)row * 6u + (unsigned)m] = c[r];
            }
        }
    }
}

// ---------------------------------------------------------------- layer2 edge scatter: 8 lanes per edge (6 active)
__global__ void __launch_bounds__(256) k_edge2(const int* __restrict__ src,
                                               const int* __restrict__ dst,
                                               const float* __restrict__ dinv,
                                               const float* __restrict__ h2,
                                               float* __restrict__ agg2, int E) {
    int t = blockIdx.x * blockDim.x + threadIdx.x;
    int e = t >> 3;
    unsigned f = (unsigned)(t & 7);
    if (e >= E || f >= 6u) return;
    unsigned s = (unsigned)src[e], d = (unsigned)dst[e];
    float w = dinv[s] * dinv[d];
    atomAddF(agg2 + d * 6u + f, h2[s * 6u + f] * w);
}

// ---------------------------------------------------------------- layer2 finalize: self-loop + bias + log_softmax
__global__ void __launch_bounds__(256) k_fin2(const float* __restrict__ agg2,
                                              const float* __restrict__ h2,
                                              const float* __restrict__ dinv,
                                              const float* __restrict__ b2,
                                              float* __restrict__ out, int n) {
    int i = blockIdx.x * blockDim.x + threadIdx.x;
    if (i >= n) return;
    unsigned u = (unsigned)i * 6u;
    float di = dinv[(unsigned)i];
    float w = di * di;
    float v[6];
    float mx = -1e30f;
    #pragma unroll
    for (int f = 0; f < 6; ++f) {
        v[f] = agg2[u + f] + h2[u + f] * w + b2[f];
        mx = v[f] > mx ? v[f] : mx;
    }
    float s = 0.0f;
    #pragma unroll
    for (int f = 0; f < 6; ++f) s += __expf(v[f] - mx);
    float lse = __logf(s);
    #pragma unroll
    for (int f = 0; f < 6; ++f) out[u + f] = v[f] - mx - lse;
}

// ================================================================ launcher
extern "C" void kernel_launch(void* const* d_in, const int* in_sizes, int n_in,
                              void* d_out, int out_size, void* d_ws, size_t ws_size,
                              hipStream_t stream) {
    const float* x    = (const float*)d_in[0];
    const int*   eidx = (const int*)d_in[1];
    const float* W1   = (const float*)d_in[2];
    const float* b1   = (const float*)d_in[3];
    const float* W2   = (const float*)d_in[4];
    const float* b2   = (const float*)d_in[5];

    const int nNodes = in_sizes[0] / 21;
    const int E      = in_sizes[1] / 2;
    const int* src = eidx;
    const int* dst = eidx + E;

    // workspace layout (floats): dinv[N] | buf1[16N] (h1 -> h2[6N]+agg2[6N]) | agg1[16N]
    float* w    = (float*)d_ws;
    float* dinv = w;                      // deg during accumulation
    float* h1   = w + (size_t)nNodes;
    float* agg1 = w + (size_t)17 * nNodes;
    float* h2   = h1;                     // reuse after h1 consumed
    float* agg2 = h1 + (size_t)6 * nNodes;

    const int BLK = 256;
    const int nBlkN   = (nNodes + BLK - 1) / BLK;
    const int nBlkE   = (E + BLK - 1) / BLK;
    const int nBlkE16 = (int)(((long long)E * 16 + BLK - 1) / BLK);
    const int nBlkE8  = (int)(((long long)E * 8 + BLK - 1) / BLK);
    const int nBlkNF  = (nNodes * 16 + BLK - 1) / BLK;
    const int tiles   = (nNodes + 15) / 16;
    const int wavesPerBlk = 4;
    const int nBlkG   = (tiles + wavesPerBlk - 1) / wavesPerBlk;

    // degree (implicit self-loop) and dinv
    k_set_one<<<nBlkN, BLK, 0, stream>>>(dinv, nNodes);
    hipMemsetAsync(agg1, 0, (size_t)16 * nNodes * sizeof(float), stream);
    k_degree<<<nBlkE, BLK, 0, stream>>>(dst, dinv, E);
    k_rsqrt<<<nBlkN, BLK, 0, stream>>>(dinv, nNodes);

    // layer 1
    k_gemm1<<<nBlkG, wavesPerBlk * 32, 0, stream>>>(x, W1, h1, nNodes);
    k_edge1<<<nBlkE16, BLK, 0, stream>>>(src, dst, dinv, h1, agg1, E);
    k_fin1<<<nBlkNF, BLK, 0, stream>>>(agg1, h1, dinv, b1, nNodes);

    // layer 2 (h1 buffer now free: reuse as h2 + agg2)
    hipMemsetAsync(agg2, 0, (size_t)6 * nNodes * sizeof(float), stream);
    k_gemm2<<<nBlkG, wavesPerBlk * 32, 0, stream>>>(agg1, W2, h2, nNodes);
    k_edge2<<<nBlkE8, BLK, 0, stream>>>(src, dst, dinv, h2, agg2, E);
    k_fin2<<<nBlkN, BLK, 0, stream>>>(agg2, h2, dinv, b2, (float*)d_out, nNodes);
}